// Contracter_14508399526217
// MI455X (gfx1250) — compile-verified
//
#include <hip/hip_runtime.h>

// ---------------------------------------------------------------------------
// Problem constants (from the reference)
// ---------------------------------------------------------------------------
#define ZB      50000          // edges
#define MULC    64             // channel multiplicity
#define BASE    9              // 0e+1o+2e -> 1+3+5
#define NPATH   11
#define ROWSZ   (MULC * BASE)  // 576 floats per z-row
#define KPAD    84             // 81 (i,j) pairs padded to multiple of 4
#define NKSTEP  (KPAD / 4)     // 21 WMMA K-steps
// B-panel per u: NKSTEP * 4 rows * 16 cols floats = 1344 floats
#define BPANEL  (NKSTEP * 4 * 16)

typedef float v2f __attribute__((ext_vector_type(2)));
typedef float v8f __attribute__((ext_vector_type(8)));

// ---------------------------------------------------------------------------
// Kernel 1: fold weights into CG tensor and pack WMMA-ready B panels.
// Bbuf layout (float): index = u*1344 + kk*64 + h*32 + col*2 + e
//   which stores B[K = 4*kk + 2*h + e, N = col] for u's 84x16 B matrix,
//   B[c, k] = sum_p w[u,p] * w3j[p, k, i, j]  with c = i*9 + j  (zero padded).
// A lane (half h, col = lane&15) then fetches its two B VGPRs as one float2.
// ---------------------------------------------------------------------------
__global__ __launch_bounds__(256) void fold_pack_kernel(
    const float* __restrict__ w,     // (64, 11)
    const float* __restrict__ w3j,   // (11, 9, 9, 9)  [p, k, i, j]
    float* __restrict__ Bbuf)        // (64, 1344)
{
    const int u = blockIdx.x;        // 0..63
    for (int t = threadIdx.x; t < BPANEL; t += blockDim.x) {
        const int e   = t & 1;
        const int col = (t >> 1) & 15;
        const int h   = (t >> 5) & 1;
        const int kk  = t >> 6;                 // 0..20
        const int c   = 4 * kk + 2 * h + e;     // K index in padded 84
        float val = 0.0f;
        if (col < BASE && c < BASE * BASE) {
            const int i = c / BASE;
            const int j = c % BASE;
            #pragma unroll
            for (int p = 0; p < NPATH; ++p) {
                val += w[u * NPATH + p] *
                       w3j[((p * BASE + col) * BASE + i) * BASE + j];
            }
        }
        Bbuf[(size_t)u * BPANEL + t] = val;
    }
}

// ---------------------------------------------------------------------------
// Kernel 2: streaming bilinear contraction via V_WMMA_F32_16X16X4_F32.
// One wave32 per task; task = ztile*64 + u covers 16 consecutive z for one u.
//   A[M=z_local, K=c]   = x1[z,u,i] * x2[z,u,j]   (formed in registers)
//   B[K=c,      N=k]    = packed panel (float2 per lane per step)
//   D[M,N] accumulated over 21 K-steps -> out[z, u, k], k < 9.
// A-frag ISA layout: lanes 0-15 M=0..15 {VGPR0:K=0, VGPR1:K=1},
//                    lanes 16-31       {VGPR0:K=2, VGPR1:K=3}  => K = 2h + e.
// B-frag ISA layout: VGPR v, half h => K = v + 2h, N = lane & 15.
// C/D layout:        VGPR r, half h => M = r + 8h, N = lane & 15.
// ---------------------------------------------------------------------------
__global__ __launch_bounds__(256) void tp_wmma_kernel(
    const float* __restrict__ x1,
    const float* __restrict__ x2,
    const float* __restrict__ Bbuf,
    float* __restrict__ out)
{
    const int lane = threadIdx.x & 31;
    const int wave = threadIdx.x >> 5;
    const int task = blockIdx.x * 8 + wave;     // 25000 blocks * 8 waves
    const int u  = task & 63;
    const int zt = task >> 6;                   // 0..3124

    const int m = lane & 15;                    // z row within tile / out col
    const int h = lane >> 4;                    // lane half

    // --- load this lane's z-row slices of x1/x2 (9 floats each) -------------
    const size_t zrow = ((size_t)zt * 16 + m) * ROWSZ + (size_t)u * BASE;
    float a1[BASE], a2[BASE];
    #pragma unroll
    for (int i = 0; i < BASE; ++i) {
        a1[i] = x1[zrow + i];
        a2[i] = x2[zrow + i];
    }

    // y(c) = x1_i * x2_j for c = i*9 + j, zero in the K padding.
    auto y = [&](int c) -> float {
        return (c < BASE * BASE) ? a1[c / BASE] * a2[c % BASE] : 0.0f;
    };

    const float* bp = Bbuf + (size_t)u * BPANEL;

    v8f acc = {0.f, 0.f, 0.f, 0.f, 0.f, 0.f, 0.f, 0.f};

    #pragma unroll
    for (int kk = 0; kk < NKSTEP; ++kk) {
        // A fragment: K_local = 2h + e  -> global c = 4kk + 2h + e
        const float a0lo = y(4 * kk + 0);   // h=0, e=0
        const float a0hi = y(4 * kk + 2);   // h=1, e=0
        const float a1lo = y(4 * kk + 1);   // h=0, e=1
        const float a1hi = y(4 * kk + 3);   // h=1, e=1
        v2f A;
        A.x = h ? a0hi : a0lo;
        A.y = h ? a1hi : a1lo;

        // B fragment: one float2 per lane = {B[4kk+2h, m], B[4kk+2h+1, m]}
        const v2f B = *(const v2f*)(bp + (size_t)kk * 64 + h * 32 + m * 2);

        acc = __builtin_amdgcn_wmma_f32_16x16x4_f32(
            /*neg_a=*/false, A, /*neg_b=*/false, B,
            /*c_mod=*/(short)0, acc, /*reuse_a=*/false, /*reuse_b=*/false);
    }

    // --- store D: row M = r + 8h, col N = m (only k < 9 is real) ------------
    if (m < BASE) {
        #pragma unroll
        for (int r = 0; r < 8; ++r) {
            const size_t z = (size_t)zt * 16 + r + 8 * h;
            out[z * ROWSZ + (size_t)u * BASE + m] = acc[r];
        }
    }
}

// ---------------------------------------------------------------------------
// Launch: inputs in setup_inputs() order: x1, x2, weights, w3j.
// d_ws holds the 344 KB packed B panels (fully overwritten each call).
// ---------------------------------------------------------------------------
extern "C" void kernel_launch(void* const* d_in, const int* in_sizes, int n_in,
                              void* d_out, int out_size, void* d_ws, size_t ws_size,
                              hipStream_t stream) {
    const float* x1      = (const float*)d_in[0];
    const float* x2      = (const float*)d_in[1];
    const float* weights = (const float*)d_in[2];
    const float* w3j     = (const float*)d_in[3];
    float*       out     = (float*)d_out;
    float*       Bbuf    = (float*)d_ws;   // 64 * 1344 floats = 344064 B

    // Fold weights into CG tensor, pack WMMA B panels (tiny, ~1M FMA).
    fold_pack_kernel<<<MULC, 256, 0, stream>>>(weights, w3j, Bbuf);

    // Main streaming contraction: 3125 z-tiles * 64 u = 200000 waves,
    // 8 waves (256 threads) per block -> 25000 blocks.
    tp_wmma_kernel<<<(ZB / 16) * MULC / 8, 256, 0, stream>>>(x1, x2, Bbuf, out);
}